// RSCNNSAModuleMSG_13967233646747
// MI455X (gfx1250) — compile-verified
//
#include <hip/hip_runtime.h>
#include <math.h>

#define B_      16
#define N_      4096
#define NPOINT_ 1024
#define CFEAT_  64
#define CIN_    67
#define CMID_   32
#define COUT_   128
#define CPAD_   80
#define NS_     16384   // B*NPOINT
#define EPS_    1e-5f

typedef __attribute__((ext_vector_type(2))) float v2f;
typedef __attribute__((ext_vector_type(8))) float v8f;

// ---------------- workspace layout (bytes) ----------------
#define OFF_NEWXYZ    65536u                          // 16*1024*3 f32
#define OFF_IDX0      262144u                         // 16*1024*32 i32
#define OFF_IDX1      2359296u                        // 16*1024*64 i32
#define OFF_GSTAT(s)  (6553600u + (unsigned)(s)*2048u)  // 65 doubles
#define OFF_W1F(s)    (6557696u + (unsigned)(s)*2048u)  // 352 f32 (folded conv1)
#define OFF_ZSTAT(s)  (6561792u + (unsigned)(s)*2048u)  // 160 doubles
#define OFF_BN2(s)    (6565888u + (unsigned)(s)*1024u)  // 160 f32
#define OFF_BN3(s)    (6567936u + (unsigned)(s)*1024u)  // 256 f32
#define OFF_Y(s)      (6569984u + (unsigned)(s)*5242880u)  // 80*16384 f32
#define OFF_U(s)      (17055744u + (unsigned)(s)*8388608u) // 128*16384 f32

// ---------------- helpers ----------------
__device__ __forceinline__ float bperm_f(int srclane, float v) {
  return __int_as_float(__builtin_amdgcn_ds_bpermute(srclane << 2, __float_as_int(v)));
}
__device__ __forceinline__ int bperm_i(int srclane, int v) {
  return __builtin_amdgcn_ds_bpermute(srclane << 2, v);
}
__device__ __forceinline__ v8f wmma4(v2f a, v2f b, v8f c) {
  // V_WMMA_F32_16X16X4_F32 : D = A(16x4 f32) * B(4x16 f32) + C(16x16 f32)
  return __builtin_amdgcn_wmma_f32_16x16x4_f32(false, a, false, b, (short)0, c, false, false);
}

// ---------------- 0: zero stats ----------------
__global__ void zero_kernel(float* p, int n) {
  int i = blockIdx.x * blockDim.x + threadIdx.x;
  if (i < n) p[i] = 0.f;
}

// ---------------- 1: furthest point sampling ----------------
__global__ void __launch_bounds__(256) fps_kernel(const float* __restrict__ xyz,
                                                  float* __restrict__ new_xyz) {
  int b = blockIdx.x;
  int tid = threadIdx.x;
  const float* x3 = xyz + (size_t)b * N_ * 3;
  float px[16], py[16], pz[16], dd[16];
#pragma unroll
  for (int j = 0; j < 16; j++) {
    int p = tid + 256 * j;
    px[j] = x3[p * 3 + 0];
    py[j] = x3[p * 3 + 1];
    pz[j] = x3[p * 3 + 2];
    dd[j] = 1e10f;
  }
  __shared__ float sP[3];
  __shared__ int s_last;
  __shared__ float wv[8];
  __shared__ int wi[8];
  if (tid == 0) s_last = 0;
  __syncthreads();
  int lane = tid & 31, wid = tid >> 5;
  for (int i = 0; i < NPOINT_; i++) {
    if (tid == 0) {
      int last = s_last;
      float lx = x3[last * 3 + 0], ly = x3[last * 3 + 1], lz = x3[last * 3 + 2];
      size_t o = ((size_t)b * NPOINT_ + i) * 3;
      new_xyz[o + 0] = lx; new_xyz[o + 1] = ly; new_xyz[o + 2] = lz;
      sP[0] = lx; sP[1] = ly; sP[2] = lz;
    }
    __syncthreads();
    float lx = sP[0], ly = sP[1], lz = sP[2];
    float bv = -1.f; int bi = 0x7fffffff;
#pragma unroll
    for (int j = 0; j < 16; j++) {
      float dx = px[j] - lx, dy = py[j] - ly, dz = pz[j] - lz;
      float d2 = dx * dx + dy * dy + dz * dz;
      dd[j] = fminf(dd[j], d2);
      int p = tid + 256 * j;
      if (dd[j] > bv || (dd[j] == bv && p < bi)) { bv = dd[j]; bi = p; }
    }
#pragma unroll
    for (int m = 1; m < 32; m <<= 1) {
      float ov = bperm_f(lane ^ m, bv);
      int   oi = bperm_i(lane ^ m, bi);
      if (ov > bv || (ov == bv && oi < bi)) { bv = ov; bi = oi; }
    }
    if (lane == 0) { wv[wid] = bv; wi[wid] = bi; }
    __syncthreads();
    if (tid == 0) {
      float fv = wv[0]; int fi = wi[0];
      for (int w = 1; w < 8; w++)
        if (wv[w] > fv || (wv[w] == fv && wi[w] < fi)) { fv = wv[w]; fi = wi[w]; }
      s_last = fi;
    }
    __syncthreads();
  }
}

// ---------------- 2: ball query (first-nsample-in-order, pad with first) ------
__global__ void ball_query_kernel(const float* __restrict__ xyz,
                                  const float* __restrict__ new_xyz,
                                  int* __restrict__ idx, float r2, int nsample) {
  int q = blockIdx.x * blockDim.x + threadIdx.x;
  if (q >= NS_) return;
  int b = q >> 10;
  const float* x3 = xyz + (size_t)b * N_ * 3;
  float qx = new_xyz[q * 3 + 0], qy = new_xyz[q * 3 + 1], qz = new_xyz[q * 3 + 2];
  int cnt = 0, first = 0;
  bool have = false;
  for (int j = 0; j < N_ && cnt < nsample; j++) {
    float dx = x3[j * 3 + 0] - qx, dy = x3[j * 3 + 1] - qy, dz = x3[j * 3 + 2] - qz;
    if (dx * dx + dy * dy + dz * dz < r2) {
      if (!have) { first = j; have = true; }
      idx[(size_t)q * nsample + cnt] = j;
      cnt++;
    }
  }
  for (int p = cnt; p < nsample; p++) idx[(size_t)q * nsample + p] = first;
}

// ---------------- 3: 10-dim geometric feature moment stats ----------------
__global__ void __launch_bounds__(256) geo_stats_kernel(
    const float* __restrict__ xyz, const float* __restrict__ new_xyz,
    const int* __restrict__ idx, int S, double* __restrict__ gstat, int nelem) {
  float acc[65];
#pragma unroll
  for (int i = 0; i < 65; i++) acc[i] = 0.f;
  int stride = gridDim.x * blockDim.x;
  for (int e = blockIdx.x * blockDim.x + threadIdx.x; e < nelem; e += stride) {
    int q = e / S, s = e - q * S;
    int b = q >> 10;
    const float* x3 = xyz + (size_t)b * N_ * 3;
    int id = idx[(size_t)q * S + s];
    int id0 = idx[(size_t)q * S];
    float qx = new_xyz[q * 3 + 0], qy = new_xyz[q * 3 + 1], qz = new_xyz[q * 3 + 2];
    float gx = x3[id * 3 + 0], gy = x3[id * 3 + 1], gz = x3[id * 3 + 2];
    float cx = x3[id0 * 3 + 0], cy = x3[id0 * 3 + 1], cz = x3[id0 * 3 + 2];
    float dx = gx - qx, dy = gy - qy, dz = gz - qz;
    float dist = sqrtf(dx * dx + dy * dy + dz * dz + 1e-12f);
    float h[10] = {dist, cx, cy, cz, gx, gy, gz, dx, dy, dz};
#pragma unroll
    for (int i = 0; i < 10; i++) acc[i] += h[i];
    int p = 10;
#pragma unroll
    for (int i = 0; i < 10; i++) {
#pragma unroll
      for (int j = i; j < 10; j++) { acc[p] += h[i] * h[j]; p++; }
    }
  }
  __shared__ float sacc[65];
  if (threadIdx.x < 65) sacc[threadIdx.x] = 0.f;
  __syncthreads();
#pragma unroll
  for (int i = 0; i < 65; i++) atomicAdd(&sacc[i], acc[i]);
  __syncthreads();
  if (threadIdx.x < 65) atomicAdd(&gstat[threadIdx.x], (double)sacc[threadIdx.x]);
}

// ---------------- 4: fold BN1 into conv1 weights (analytic mean/var) ---------
__global__ void finalize_bn1_kernel(const double* __restrict__ gstat,
                                    const float* __restrict__ w1, const float* __restrict__ b1,
                                    const float* __restrict__ g, const float* __restrict__ be,
                                    float* __restrict__ w1f, double inv) {
  __shared__ double smu[10];
  __shared__ double scov[100];
  if (threadIdx.x == 0) {
    for (int i = 0; i < 10; i++) smu[i] = gstat[i] * inv;
    int p = 10;
    for (int i = 0; i < 10; i++)
      for (int j = i; j < 10; j++) {
        double c = gstat[p++] * inv - smu[i] * smu[j];
        scov[i * 10 + j] = c; scov[j * 10 + i] = c;
      }
  }
  __syncthreads();
  int o = threadIdx.x;
  if (o < CMID_) {
    double mo = (double)b1[o];
    for (int j = 0; j < 10; j++) mo += (double)w1[o * 10 + j] * smu[j];
    double vo = 0.0;
    for (int j = 0; j < 10; j++)
      for (int k = 0; k < 10; k++)
        vo += (double)w1[o * 10 + j] * (double)w1[o * 10 + k] * scov[j * 10 + k];
    float a = g[o] / sqrtf((float)vo + EPS_);
    for (int j = 0; j < 10; j++) w1f[o * 10 + j] = w1[o * 10 + j] * a;
    w1f[320 + o] = (float)((double)b1[o] - mo) * a + be[o];
  }
}

// ---------------- 5/7: fused conv1+conv2(WMMA)+z  (MODE 0: stats, 1: bn+max) -
template <int MODE>
__global__ void __launch_bounds__(256) z_pass_kernel(
    const float* __restrict__ xyz, const float* __restrict__ new_xyz,
    const int* __restrict__ idx, const float* __restrict__ w2,
    const float* __restrict__ b2, const float* __restrict__ feat,
    const float* __restrict__ w1f, const float* __restrict__ bn2,
    double* __restrict__ zstat, float* __restrict__ y, int S) {
  __shared__ float s_w1f[352];
  __shared__ float s_w2[CPAD_ * CMID_];
  __shared__ float s_b2[CPAD_];
  __shared__ float s_a[CPAD_], s_c[CPAD_];
  int tid = threadIdx.x;
  for (int i = tid; i < 352; i += 256) s_w1f[i] = w1f[i];
  for (int i = tid; i < CPAD_ * CMID_; i += 256) {
    int c = i >> 5;
    s_w2[i] = (c < CIN_) ? w2[c * CMID_ + (i & 31)] : 0.f;
  }
  for (int i = tid; i < CPAD_; i += 256) s_b2[i] = (i < CIN_) ? b2[i] : 0.f;
  if (MODE == 1) {
    for (int i = tid; i < CPAD_; i += 256) { s_a[i] = bn2[i]; s_c[i] = bn2[CPAD_ + i]; }
  }
  __syncthreads();

  int lane = tid & 31;
  int hs = lane >> 4;     // half-wave selector
  int ls = lane & 15;     // sample / M index within tile
  int base = hs * 2;      // K sub-offset for A/B operands
  int gw = (blockIdx.x * blockDim.x + tid) >> 5;
  int nwaves = (gridDim.x * blockDim.x) >> 5;
  int ntile = S >> 4;

  float ssum[40], ssq[40];
  if (MODE == 0) {
#pragma unroll
    for (int i = 0; i < 40; i++) { ssum[i] = 0.f; ssq[i] = 0.f; }
  }

  for (int q = gw; q < NS_; q += nwaves) {
    int b = q >> 10;
    const float* x3 = xyz + (size_t)b * N_ * 3;
    float qx = new_xyz[q * 3 + 0], qy = new_xyz[q * 3 + 1], qz = new_xyz[q * 3 + 2];
    int id0 = idx[(size_t)q * S];
    float cx = x3[id0 * 3 + 0], cy = x3[id0 * 3 + 1], cz = x3[id0 * 3 + 2];
    float vmax[40];
    if (MODE == 1) {
#pragma unroll
      for (int i = 0; i < 40; i++) vmax[i] = -1e30f;
    }
    for (int t = 0; t < ntile; t++) {
      int s = t * 16 + ls;
      int id = idx[(size_t)q * S + s];
      float gx = x3[id * 3 + 0], gy = x3[id * 3 + 1], gz = x3[id * 3 + 2];
      float dx = gx - qx, dy = gy - qy, dz = gz - qz;
      float dist = sqrtf(dx * dx + dy * dy + dz * dz + 1e-12f);
      float h[10] = {dist, cx, cy, cz, gx, gy, gz, dx, dy, dz};
      // Build B operand: H1[32 x 16] = relu(folded_conv1(h)), per-lane K pairs
      v2f bb[8];
#pragma unroll
      for (int i = 0; i < 8; i++) {
        int c0 = 4 * i + base;
        float v0 = s_w1f[320 + c0], v1 = s_w1f[320 + c0 + 1];
#pragma unroll
        for (int j = 0; j < 10; j++) {
          v0 += s_w1f[c0 * 10 + j] * h[j];
          v1 += s_w1f[(c0 + 1) * 10 + j] * h[j];
        }
        bb[i].x = fmaxf(v0, 0.f);
        bb[i].y = fmaxf(v1, 0.f);
      }
      // H2 = W2(80x32) * H1 via 5 row-tiles x 8 K-steps of wmma f32 16x16x4
#pragma unroll
      for (int rt = 0; rt < 5; rt++) {
        v8f acc;
#pragma unroll
        for (int r = 0; r < 8; r++) acc[r] = s_b2[rt * 16 + r + hs * 8];
        int arow = rt * 16 + ls;
#pragma unroll
        for (int i = 0; i < 8; i++) {
          v2f aa;
          aa.x = s_w2[arow * 32 + 4 * i + base];
          aa.y = s_w2[arow * 32 + 4 * i + base + 1];
          acc = wmma4(aa, bb[i], acc);
        }
#pragma unroll
        for (int r = 0; r < 8; r++) {
          int o = rt * 16 + r + hs * 8;
          float xv;
          if (o == 0) xv = dx;
          else if (o == 1) xv = dy;
          else if (o == 2) xv = dz;
          else if (o < CIN_) xv = feat[((size_t)b * N_ + id) * CFEAT_ + (o - 3)];
          else xv = 0.f;
          float z = acc[r] * xv;
          int ai = rt * 8 + r;
          if (MODE == 0) {
            ssum[ai] += z; ssq[ai] += z * z;
          } else {
            float zz = fmaxf(z * s_a[o] + s_c[o], 0.f);
            vmax[ai] = fmaxf(vmax[ai], zz);
          }
        }
      }
    }
    if (MODE == 1) {
#pragma unroll
      for (int i = 0; i < 40; i++) {
        float v = vmax[i];
#pragma unroll
        for (int m = 1; m < 16; m <<= 1) v = fmaxf(v, bperm_f(lane ^ m, v));
        vmax[i] = v;
      }
      if (ls == 0) {
#pragma unroll
        for (int rt = 0; rt < 5; rt++)
#pragma unroll
          for (int r = 0; r < 8; r++) {
            int o = rt * 16 + r + hs * 8;
            y[(size_t)o * NS_ + q] = vmax[rt * 8 + r];
          }
      }
    }
  }
  if (MODE == 0) {
#pragma unroll
    for (int i = 0; i < 40; i++) {
      float a = ssum[i], s2 = ssq[i];
#pragma unroll
      for (int m = 1; m < 16; m <<= 1) {
        a += bperm_f(lane ^ m, a);
        s2 += bperm_f(lane ^ m, s2);
      }
      ssum[i] = a; ssq[i] = s2;
    }
    if (ls == 0) {
      for (int rt = 0; rt < 5; rt++)
        for (int r = 0; r < 8; r++) {
          int o = rt * 16 + r + hs * 8;
          int ai = rt * 8 + r;
          if (o < CIN_) {
            atomicAdd(&zstat[o], (double)ssum[ai]);
            atomicAdd(&zstat[CPAD_ + o], (double)ssq[ai]);
          }
        }
    }
  }
}

// ---------------- 6: fold BN2 ----------------
__global__ void finalize_bn2_kernel(const double* __restrict__ zstat,
                                    const float* __restrict__ g, const float* __restrict__ be,
                                    float* __restrict__ bn2, double inv) {
  int c = threadIdx.x;
  if (c >= CPAD_) return;
  if (c < CIN_) {
    float mean = (float)(zstat[c] * inv);
    float var = (float)(zstat[CPAD_ + c] * inv) - mean * mean;
    float a = g[c] / sqrtf(var + EPS_);
    bn2[c] = a;
    bn2[CPAD_ + c] = be[c] - mean * a;
  } else {
    bn2[c] = 0.f;
    bn2[CPAD_ + c] = 0.f;
  }
}

// ---------------- 8: u = W_cr(128x80) * y(80x16384) via WMMA -----------------
__global__ void __launch_bounds__(256) gemm_cr_kernel(const float* __restrict__ wcr,
                                                      const float* __restrict__ bcr,
                                                      const float* __restrict__ y,
                                                      float* __restrict__ u) {
  __shared__ float s_w[COUT_ * CPAD_];
  __shared__ float s_b[COUT_];
  int tid = threadIdx.x;
  for (int i = tid; i < COUT_ * CPAD_; i += 256) {
    int o = i / CPAD_, k = i - o * CPAD_;
    s_w[i] = (k < CIN_) ? wcr[o * CIN_ + k] : 0.f;
  }
  for (int i = tid; i < COUT_; i += 256) s_b[i] = bcr[i];
  __syncthreads();
  int lane = tid & 31, hs = lane >> 4, ls = lane & 15, base = hs * 2;
  int gw = (blockIdx.x * blockDim.x + tid) >> 5;
  int nw = (gridDim.x * blockDim.x) >> 5;
  for (int st = gw; st < NS_ / 16; st += nw) {
    int s0 = st * 16;
    v2f bb[20];
#pragma unroll
    for (int ks = 0; ks < 20; ks++) {
      int k0 = 4 * ks + base;
      bb[ks].x = y[(size_t)k0 * NS_ + s0 + ls];
      bb[ks].y = y[(size_t)(k0 + 1) * NS_ + s0 + ls];
    }
#pragma unroll
    for (int rt = 0; rt < 8; rt++) {
      v8f acc;
#pragma unroll
      for (int r = 0; r < 8; r++) acc[r] = s_b[rt * 16 + r + hs * 8];
      int arow = rt * 16 + ls;
#pragma unroll
      for (int ks = 0; ks < 20; ks++) {
        v2f aa;
        aa.x = s_w[arow * CPAD_ + 4 * ks + base];
        aa.y = s_w[arow * CPAD_ + 4 * ks + base + 1];
        acc = wmma4(aa, bb[ks], acc);
      }
#pragma unroll
      for (int r = 0; r < 8; r++) {
        int o = rt * 16 + r + hs * 8;
        u[(size_t)o * NS_ + s0 + ls] = acc[r];
      }
    }
  }
}

// ---------------- 9: per-channel stats of u -> BN3 params --------------------
__global__ void __launch_bounds__(256) u_stats_kernel(const float* __restrict__ u,
                                                      const float* __restrict__ g,
                                                      const float* __restrict__ be,
                                                      float* __restrict__ bn3) {
  int c = blockIdx.x;
  const float* row = u + (size_t)c * NS_;
  float s = 0.f, s2 = 0.f;
  for (int i = threadIdx.x; i < NS_; i += blockDim.x) {
    float v = row[i];
    s += v; s2 += v * v;
  }
  __shared__ float sh[256], sh2[256];
  sh[threadIdx.x] = s; sh2[threadIdx.x] = s2;
  __syncthreads();
  for (int off = 128; off; off >>= 1) {
    if (threadIdx.x < off) {
      sh[threadIdx.x] += sh[threadIdx.x + off];
      sh2[threadIdx.x] += sh2[threadIdx.x + off];
    }
    __syncthreads();
  }
  if (threadIdx.x == 0) {
    float mean = sh[0] / (float)NS_;
    float var = sh2[0] / (float)NS_ - mean * mean;
    float a = g[c] / sqrtf(var + EPS_);
    bn3[c] = a;
    bn3[COUT_ + c] = be[c] - mean * a;
  }
}

// ---------------- 10: final relu(bn3(u)) concat write ------------------------
__global__ void final_out_kernel(const float* __restrict__ u0, const float* __restrict__ u1,
                                 const float* __restrict__ bn30, const float* __restrict__ bn31,
                                 float* __restrict__ out) {
  int e = blockIdx.x * blockDim.x + threadIdx.x;
  if (e >= B_ * 256 * NPOINT_) return;
  int m = e & 1023;
  int c = (e >> 10) & 255;
  int b = e >> 18;
  int sc = c >> 7, cc = c & 127;
  const float* u = sc ? u1 : u0;
  const float* bn = sc ? bn31 : bn30;
  float v = u[(size_t)cc * NS_ + b * NPOINT_ + m];
  out[e] = fmaxf(v * bn[cc] + bn[COUT_ + cc], 0.f);
}

// ---------------- host launch ----------------
extern "C" void kernel_launch(void* const* d_in, const int* in_sizes, int n_in,
                              void* d_out, int out_size, void* d_ws, size_t ws_size,
                              hipStream_t stream) {
  (void)in_sizes; (void)n_in; (void)out_size; (void)ws_size;
  const float* xyz   = (const float*)d_in[0];
  const float* feat  = (const float*)d_in[2];   // normal (d_in[1]) unused by reference
  const float* w1    = (const float*)d_in[3];
  const float* b1    = (const float*)d_in[4];
  const float* w2    = (const float*)d_in[5];
  const float* b2    = (const float*)d_in[6];
  const float* wcr   = (const float*)d_in[7];
  const float* bcr   = (const float*)d_in[8];
  const float* gmap  = (const float*)d_in[9];
  const float* bemap = (const float*)d_in[10];
  const float* grs   = (const float*)d_in[11];
  const float* bers  = (const float*)d_in[12];
  const float* gcr   = (const float*)d_in[13];
  const float* becr  = (const float*)d_in[14];
  char* ws = (char*)d_ws;
  float* out = (float*)d_out;
  float* newxyz = (float*)(ws + OFF_NEWXYZ);

  // zero GSTAT/W1F/ZSTAT region (12288 bytes = 3072 f32)
  zero_kernel<<<(3072 + 255) / 256, 256, 0, stream>>>((float*)(ws + OFF_GSTAT(0)), 3072);
  fps_kernel<<<B_, 256, 0, stream>>>(xyz, newxyz);

  for (int s = 0; s < 2; s++) {
    int S = s ? 64 : 32;
    float r = s ? 0.2f : 0.1f;
    int* idx      = (int*)(ws + (s ? OFF_IDX1 : OFF_IDX0));
    double* gstat = (double*)(ws + OFF_GSTAT(s));
    float* w1f    = (float*)(ws + OFF_W1F(s));
    double* zstat = (double*)(ws + OFF_ZSTAT(s));
    float* bn2    = (float*)(ws + OFF_BN2(s));
    float* bn3    = (float*)(ws + OFF_BN3(s));
    float* y      = (float*)(ws + OFF_Y(s));
    float* u      = (float*)(ws + OFF_U(s));
    int nelem = NS_ * S;
    double inv = 1.0 / (double)nelem;

    ball_query_kernel<<<NS_ / 256, 256, 0, stream>>>(xyz, newxyz, idx, r * r, S);
    geo_stats_kernel<<<512, 256, 0, stream>>>(xyz, newxyz, idx, S, gstat, nelem);
    finalize_bn1_kernel<<<1, 32, 0, stream>>>(gstat, w1, b1, gmap, bemap, w1f, inv);
    z_pass_kernel<0><<<256, 256, 0, stream>>>(xyz, newxyz, idx, w2, b2, feat, w1f, bn2,
                                              zstat, y, S);
    finalize_bn2_kernel<<<1, 128, 0, stream>>>(zstat, grs, bers, bn2, inv);
    z_pass_kernel<1><<<2048, 256, 0, stream>>>(xyz, newxyz, idx, w2, b2, feat, w1f, bn2,
                                               zstat, y, S);
    gemm_cr_kernel<<<128, 256, 0, stream>>>(wcr, bcr, y, u);
    u_stats_kernel<<<COUT_, 256, 0, stream>>>(u, gcr, becr, bn3);
  }

  final_out_kernel<<<(B_ * 256 * NPOINT_ + 255) / 256, 256, 0, stream>>>(
      (const float*)(ws + OFF_U(0)), (const float*)(ws + OFF_U(1)),
      (const float*)(ws + OFF_BN3(0)), (const float*)(ws + OFF_BN3(1)), out);
}